// MaskedConv2dDynamicDW_12884901888106
// MI455X (gfx1250) — compile-verified
//
#include <hip/hip_runtime.h>
#include <hip/hip_bf16.h>

// ---------------------------------------------------------------------------
// Problem constants
// ---------------------------------------------------------------------------
#define DIMC   192
#define REDC   48          // DIM / RED
#define HH     256
#define WW     256
#define PLANE  (HH * WW)   // 65536
#define NB     4
#define KK9    1728        // DIM * 3 * 3
#define EPSBN  1e-5f

typedef __attribute__((ext_vector_type(2))) float v2f;
typedef __attribute__((ext_vector_type(8))) float v8f;

// ---------------------------------------------------------------------------
// Kernel 1: global average pool per (b, c) plane.  768 blocks x 256 threads.
// Each block reduces 65536 floats (16384 float4 loads).
// ---------------------------------------------------------------------------
__global__ __launch_bounds__(256) void pool_kernel(const float* __restrict__ x,
                                                   float* __restrict__ pooled) {
    __shared__ float red[256];
    const size_t base = (size_t)blockIdx.x * PLANE;
    const float4* p = (const float4*)(x + base);
    float s = 0.0f;
    for (int i = threadIdx.x; i < PLANE / 4; i += 256) {
        float4 v = p[i];
        s += v.x + v.y + v.z + v.w;
    }
    red[threadIdx.x] = s;
    __syncthreads();
    for (int off = 128; off > 0; off >>= 1) {
        if (threadIdx.x < off) red[threadIdx.x] += red[threadIdx.x + off];
        __syncthreads();
    }
    if (threadIdx.x == 0) pooled[blockIdx.x] = red[0] * (1.0f / (float)PLANE);
}

// ---------------------------------------------------------------------------
// Kernel 2: t = ReLU(BN(pooled @ w1^T)).  4*48 = 192 outputs, one thread each.
// ---------------------------------------------------------------------------
__global__ __launch_bounds__(192) void gen1_kernel(const float* __restrict__ pooled,
                                                   const float* __restrict__ w1,
                                                   const float* __restrict__ gamma,
                                                   const float* __restrict__ beta,
                                                   const float* __restrict__ rmean,
                                                   const float* __restrict__ rvar,
                                                   float* __restrict__ tout) {
    const int idx = threadIdx.x;       // = b*48 + r
    const int b = idx / REDC;
    const int r = idx % REDC;
    float s = 0.0f;
    #pragma unroll 4
    for (int k = 0; k < DIMC; ++k)
        s = fmaf(pooled[b * DIMC + k], w1[r * DIMC + k], s);
    float v = gamma[r] * (s - rmean[r]) * rsqrtf(rvar[r] + EPSBN) + beta[r];
    tout[idx] = fmaxf(v, 0.0f);
}

// ---------------------------------------------------------------------------
// Kernel 3: wt = t @ w2^T + b2, masked, via V_WMMA_F32_16X16X4_F32.
//   A = t  (M=16 rows, only 0..3 valid, K=48 in chunks of 4)
//   B = w2^T chunk (K=4 x N=16), w2 is [1728][48] row-major
//   D = [16x16] f32; rows 0..3 are wt[b][n0+lane].
// Layout per ISA 7.12.2 (32-bit A 16x4): lanes 0-15 hold K={k0,k0+1} in
// vgpr{0,1}; lanes 16-31 hold K={k0+2,k0+3}. B mirrors (rows striped across
// lanes). One wave per 16-column tile: 108 blocks x 32 threads.
//
// A-row padding (M=4..15 must be zero) is done with a CLAMPED index plus a
// 0/1 lane-mask multiply instead of a predicated load, so the loads stay
// straight-line (no EXEC save/restore branches around each global_load).
// ---------------------------------------------------------------------------
__global__ __launch_bounds__(32) void gen2_wmma_kernel(const float* __restrict__ t,
                                                       const float* __restrict__ w2,
                                                       const float* __restrict__ b2,
                                                       float* __restrict__ wgen) {
    const int lane = threadIdx.x;       // 0..31
    const int half = lane >> 4;         // 0: K pair {0,1}; 1: K pair {2,3}
    const int l16  = lane & 15;
    const int n0   = blockIdx.x * 16;
    const int col  = n0 + l16;          // N index (0..1727)

    const int   mrow = (l16 < NB) ? l16 : 0;     // clamped A row (always valid)
    const float amsk = (l16 < NB) ? 1.0f : 0.0f; // zero out padding rows
    const float* __restrict__ arow = t  + mrow * REDC;
    const float* __restrict__ brow = w2 + col  * REDC;

    v8f acc = {};
    #pragma unroll
    for (int k0 = 0; k0 < REDC; k0 += 4) {
        const int ka = k0 + 2 * half;
        v2f a, bm;
        // unconditional b64 loads; mask applied arithmetically
        float2 av = *(const float2*)(arow + ka);
        float2 bv = *(const float2*)(brow + ka);
        a.x  = av.x * amsk;
        a.y  = av.y * amsk;
        bm.x = bv.x;
        bm.y = bv.y;
        acc = __builtin_amdgcn_wmma_f32_16x16x4_f32(
            /*neg_a=*/false, a, /*neg_b=*/false, bm,
            /*c_mod=*/(short)0, acc, /*reuse_a=*/false, /*reuse_b=*/false);
    }

    // D layout: lanes 0-15 vgpr r hold D[M=r][N=l16]; only M=0..3 valid.
    if (half == 0) {
        const float bb  = b2[col];
        const int   tap = col % 9;            // mask 'A': keep taps 0..3 only
        const bool  keep = (tap < 4);
        #pragma unroll
        for (int r = 0; r < NB; ++r) {
            float v = acc[r] + bb;
            wgen[r * KK9 + col] = keep ? v : 0.0f;
        }
    }
}

// ---------------------------------------------------------------------------
// Kernel 4: 4-tap depthwise conv (mask 'A' leaves taps (0,0),(0,1),(0,2),(1,0))
// out(y,x) = w00*in(y-1,x-1) + w01*in(y-1,x) + w02*in(y-1,x+1)
//          + w10*in(y,  x-1) + bias
// float4 vectorized: each thread produces 4 contiguous outputs.
// Grid: (768 planes, 16 row-groups), 256 threads, 4 passes -> 16 rows/block.
// ---------------------------------------------------------------------------
__global__ __launch_bounds__(256) void dwconv_kernel(const float* __restrict__ x,
                                                     const float* __restrict__ wgen,
                                                     const float* __restrict__ bias,
                                                     float* __restrict__ out) {
    const int plane = blockIdx.x;          // b*192 + c
    const int c = plane % DIMC;
    const int b = plane / DIMC;
    const size_t base = (size_t)plane * PLANE;
    const float* __restrict__ in = x + base;
    float* __restrict__ op = out + base;

    const int wi = b * KK9 + c * 9;
    const float w00 = wgen[wi + 0];
    const float w01 = wgen[wi + 1];
    const float w02 = wgen[wi + 2];
    const float w10 = wgen[wi + 3];
    const float bc  = bias[c];

    const int tid   = threadIdx.x;
    const int xq    = (tid & 63) * 4;                 // 0,4,...,252
    const int rbase = blockIdx.y * 16 + (tid >> 6);   // +0..3

    #pragma unroll
    for (int p = 0; p < 4; ++p) {
        const int row = rbase + p * 4;
        const float* rc = in + row * WW;

        // prefetch the next row-group's data (emits global_prefetch_b8)
        if (p == 0 && row + 16 < HH)
            __builtin_prefetch(in + (row + 16) * WW + xq, 0, 1);

        float4 c0 = *(const float4*)(rc + xq);
        float  l1 = (xq > 0) ? rc[xq - 1] : 0.0f;

        float4 t0 = make_float4(0.f, 0.f, 0.f, 0.f);
        float  l0 = 0.0f, rr = 0.0f;
        if (row > 0) {
            const float* rm = rc - WW;
            t0 = *(const float4*)(rm + xq);
            l0 = (xq > 0)        ? rm[xq - 1] : 0.0f;
            rr = (xq + 4 < WW)   ? rm[xq + 4] : 0.0f;
        }

        float4 o;
        o.x = fmaf(w00, l0,   fmaf(w01, t0.x, fmaf(w02, t0.y, fmaf(w10, l1,   bc))));
        o.y = fmaf(w00, t0.x, fmaf(w01, t0.y, fmaf(w02, t0.z, fmaf(w10, c0.x, bc))));
        o.z = fmaf(w00, t0.y, fmaf(w01, t0.z, fmaf(w02, t0.w, fmaf(w10, c0.y, bc))));
        o.w = fmaf(w00, t0.z, fmaf(w01, t0.w, fmaf(w02, rr,   fmaf(w10, c0.z, bc))));

        *(float4*)(op + row * WW + xq) = o;
    }
}

// ---------------------------------------------------------------------------
// Launch
// ---------------------------------------------------------------------------
extern "C" void kernel_launch(void* const* d_in, const int* in_sizes, int n_in,
                              void* d_out, int out_size, void* d_ws, size_t ws_size,
                              hipStream_t stream) {
    const float* x     = (const float*)d_in[0];
    const float* w1    = (const float*)d_in[1];
    const float* gamma = (const float*)d_in[2];
    const float* beta  = (const float*)d_in[3];
    const float* rmean = (const float*)d_in[4];
    const float* rvar  = (const float*)d_in[5];
    const float* w2    = (const float*)d_in[6];
    const float* b2    = (const float*)d_in[7];
    const float* bias  = (const float*)d_in[8];
    float* out = (float*)d_out;

    float* ws     = (float*)d_ws;
    float* pooled = ws;              // 768 floats
    float* tbuf   = ws + 768;        // 192 floats
    float* wgen   = ws + 960;        // 6912 floats

    pool_kernel<<<NB * DIMC, 256, 0, stream>>>(x, pooled);
    gen1_kernel<<<1, 192, 0, stream>>>(pooled, w1, gamma, beta, rmean, rvar, tbuf);
    gen2_wmma_kernel<<<KK9 / 16, 32, 0, stream>>>(tbuf, w2, b2, wgen);
    dim3 cgrid(NB * DIMC, HH / 16);
    dwconv_kernel<<<cgrid, 256, 0, stream>>>(x, wgen, bias, out);
}